// TabCBM_46110768889918
// MI455X (gfx1250) — compile-verified
//
#include <hip/hip_runtime.h>
#include <cstddef>

// Problem dims (match reference)
#define B_    4096
#define D_    2048
#define C_    16
#define L_    128
#define H_    256
#define HF_   512
#define HG_   128
#define NCLS_ 200

typedef float v2f __attribute__((ext_vector_type(2)));
typedef float v8f __attribute__((ext_vector_type(8)));

#define AS1 __attribute__((address_space(1)))
#define AS3 __attribute__((address_space(3)))

#if __has_builtin(__builtin_amdgcn_global_load_async_to_lds_b32) && \
    __has_builtin(__builtin_amdgcn_s_wait_asynccnt)
#define USE_ASYNC_LDS 1
#else
#define USE_ASYNC_LDS 0
#endif

#if USE_ASYNC_LDS
// async DMA: 4B per lane, global -> LDS, tracked by ASYNCcnt
__device__ __forceinline__ void async_b32(const float* g, float* l) {
  __builtin_amdgcn_global_load_async_to_lds_b32((AS1 int*)g, (AS3 int*)l, 0, 0);
}
#endif

// ---------------------------------------------------------------------------
// Precompute gate = sigmoid(fp) and gmean = (1-gate)*mean  (C x D tables)
// so the masked-input GEMMs fold the gating into one FMA per A element
// (avoids materializing the 512 MB masked tensor).
// ---------------------------------------------------------------------------
__global__ __launch_bounds__(256) void gate_kernel(const float* __restrict__ fp,
                                                   const float* __restrict__ mean,
                                                   float* __restrict__ gate,
                                                   float* __restrict__ gmean) {
  int i = blockIdx.x * 256 + threadIdx.x;
  if (i >= C_ * D_) return;
  float g = 1.0f / (1.0f + __expf(-fp[i]));
  gate[i]  = g;
  gmean[i] = (1.0f - g) * mean[i & (D_ - 1)];
}

// ---------------------------------------------------------------------------
// fp32 WMMA GEMM:  out[c] = act(A[c] (MxK) @ B[c] (KxN) + bias[c])
// Block = 256 threads = 8 waves (wave32). Block tile 128(M) x 128(N);
// wave w owns rows [m0+16w, m0+16w+16), 8 accumulators of 16 cols each.
// B is staged through double-buffered LDS (16x128 fp32 chunks, 2 x 8 KB)
// in K-PAIR-INTERLEAVED layout  idx(r,c) = ((r>>1)*128 + c)*2 + (r&1)
// so each WMMA B fragment (B[k][n], B[k+1][n]) is one aligned ds_load_b64
// straight into an even VGPR pair (no re-pairing movs). Staging uses
// GLOBAL_LOAD_ASYNC_TO_LDS_B32 (ASYNCcnt) when available, else ds_store.
// The chunk k+1 copy is issued before the chunk-k compute so the DMA
// overlaps 32 WMMAs; one s_wait_asynccnt + one barrier per 16-K chunk.
// Fragment layouts per CDNA5 ISA 7.12.2 (V_WMMA_F32_16X16X4_F32):
//   A 16x4 : lane l, vgpr j -> A[l&15][2*(l>>4)+j]
//   B 4x16 : lane l, vgpr j -> B[2*(l>>4)+j][l&15]
//   C/D    : vgpr r, lane l -> row r+8*(l>>4), col l&15
// EXEC stays all-ones (no divergence; every dim is an exact tile multiple).
// ---------------------------------------------------------------------------
#define KC 16  // K-chunk staged per LDS buffer

__device__ __forceinline__ int lds_pair_idx(int r, int c) {
  return (((r >> 1) * 128 + c) << 1) + (r & 1);
}

template <bool MASKED>
__global__ __launch_bounds__(256) void wmma_gemm(
    const float* __restrict__ A, long long a_cs, int lda,
    const float* __restrict__ Bm, long long b_cs, int ldb,
    const float* __restrict__ bias, long long bias_cs,
    float* __restrict__ out, long long out_cs, int out_rs,
    int K,
    const float* __restrict__ gate, const float* __restrict__ gmean,
    int relu) {
  __shared__ __align__(16) float sb[2][KC * 128];

  const int c     = blockIdx.z;
  const int n0    = blockIdx.x * 128;
  const int m0    = blockIdx.y * 128;
  const int wave  = threadIdx.x >> 5;
  const int lane  = threadIdx.x & 31;
  const int lsub  = lane & 15;
  const int khalf = lane >> 4;
  const int arow  = m0 + wave * 16 + lsub;

  const float* Ab = A + (size_t)c * a_cs;
  const float* Bb = Bm + (size_t)c * b_cs;

  // Cooperative staging geometry: thread covers 8 consecutive floats of the
  // 16x128 chunk: row = tid>>4, cols 8*(tid&15) .. +7.
  const int srow = threadIdx.x >> 4;
  const int scol = (threadIdx.x & 15) * 8;
  const int sbase = lds_pair_idx(srow, scol);  // +2 per column step

  // ---- preload chunk 0 into buffer 0 ----
  {
    const float* src = Bb + (size_t)srow * ldb + n0 + scol;
#if USE_ASYNC_LDS
#pragma unroll
    for (int i = 0; i < 8; ++i) async_b32(src + i, &sb[0][sbase + 2 * i]);
    __builtin_amdgcn_s_wait_asynccnt(0);
#else
    float4 r0 = *(const float4*)src;
    float4 r1 = *(const float4*)(src + 4);
    float* dst = &sb[0][sbase];
    dst[0] = r0.x; dst[2] = r0.y; dst[4] = r0.z; dst[6] = r0.w;
    dst[8] = r1.x; dst[10] = r1.y; dst[12] = r1.z; dst[14] = r1.w;
#endif
  }
  __syncthreads();

  v8f acc[8];
#pragma unroll
  for (int t = 0; t < 8; ++t) acc[t] = v8f{};

  int p = 0;
  for (int kb = 0; kb < K; kb += KC) {
    const bool more = (kb + KC) < K;

    // 1) kick off staging of the next chunk into buffer p^1
#if !USE_ASYNC_LDS
    float4 r0, r1;
#endif
    if (more) {
      const float* src = Bb + (size_t)(kb + KC + srow) * ldb + n0 + scol;
#if USE_ASYNC_LDS
      float* dstb = &sb[p ^ 1][sbase];
#pragma unroll
      for (int i = 0; i < 8; ++i) async_b32(src + i, dstb + 2 * i);
#else
      r0 = *(const float4*)src;
      r1 = *(const float4*)(src + 4);
#endif
    }

    // 2) A fragments for the whole chunk (issued together, one wait)
    v2f af[4];
#pragma unroll
    for (int j = 0; j < 4; ++j) {
      const int kk = kb + 4 * j + 2 * khalf;
      if (MASKED) {
        const v2f g  = *(const v2f*)(gate  + (size_t)c * lda + kk);
        const v2f gm = *(const v2f*)(gmean + (size_t)c * lda + kk);
        const v2f xx = *(const v2f*)(Ab + (size_t)arow * lda + kk);
        af[j].x = fmaf(g.x, xx.x, gm.x);
        af[j].y = fmaf(g.y, xx.y, gm.y);
      } else {
        af[j] = *(const v2f*)(Ab + (size_t)arow * lda + kk);
      }
    }

    // 3) 32 WMMAs against the LDS-resident chunk; each B fragment is a
    //    single aligned ds_load_b64 thanks to the pair-interleaved layout.
    const float* bl = &sb[p][0];
#pragma unroll
    for (int j = 0; j < 4; ++j) {
      const int rel = 4 * j + 2 * khalf;  // even
      const int pbase = ((rel >> 1) * 128) << 1;
#pragma unroll
      for (int t = 0; t < 8; ++t) {
        const int col = 16 * t + lsub;
        const v2f b = *(const v2f*)(bl + pbase + 2 * col);
        acc[t] = __builtin_amdgcn_wmma_f32_16x16x4_f32(
            false, af[j], false, b, (short)0, acc[t], false, false);
      }
    }

    // 4) finish staging, publish, flip buffers
    if (more) {
#if USE_ASYNC_LDS
      __builtin_amdgcn_s_wait_asynccnt(0);
#else
      float* dst = &sb[p ^ 1][sbase];
      dst[0] = r0.x; dst[2] = r0.y; dst[4] = r0.z; dst[6] = r0.w;
      dst[8] = r1.x; dst[10] = r1.y; dst[12] = r1.z; dst[14] = r1.w;
#endif
    }
    __syncthreads();
    p ^= 1;
  }

  float* Ob = out + (size_t)c * out_cs;
#pragma unroll
  for (int t = 0; t < 8; ++t) {
    const int col = n0 + 16 * t + lsub;
    const float bv = bias ? bias[(size_t)c * bias_cs + col] : 0.0f;
#pragma unroll
    for (int r = 0; r < 8; ++r) {
      const int row = m0 + wave * 16 + 8 * khalf + r;
      float v = acc[t][r] + bv;
      if (relu) v = fmaxf(v, 0.0f);
      Ob[(size_t)row * out_rs + col] = v;
    }
  }
}

// ---------------------------------------------------------------------------
// Per-sample fusion: L2-normalize cm across the concept axis, concept_prob =
// sigmoid(<cm_norm, latent>_L), bottleneck mix of the two halves.
// cm / lat stored as (B, C, L); one block per b, thread t = L index.
// ---------------------------------------------------------------------------
__global__ __launch_bounds__(128) void fuse_kernel(const float* __restrict__ cm,
                                                   const float* __restrict__ lat,
                                                   float* __restrict__ bot) {
  __shared__ float s_cm[C_][L_];
  __shared__ float s_d[L_ * C_];
  __shared__ float s_p[C_];
  const int b = blockIdx.x;
  const int t = threadIdx.x;  // l index, 0..127
  const float* cb = cm + (size_t)b * C_ * L_ + t;
  const float* lb = lat + (size_t)b * C_ * L_ + t;

  float v[C_];
  float sumsq = 0.0f;
#pragma unroll
  for (int c = 0; c < C_; ++c) {
    v[c] = cb[c * L_];
    sumsq += v[c] * v[c];
  }
  const float inv = rsqrtf(fmaxf(sumsq, 1e-12f));
#pragma unroll
  for (int c = 0; c < C_; ++c) {
    v[c] *= inv;
    s_cm[c][t] = v[c];
    s_d[t * C_ + c] = v[c] * lb[c * L_];
  }
  __syncthreads();
  if (t < C_) {
    float s = 0.0f;
    for (int i = 0; i < L_; ++i) s += s_d[i * C_ + t];
    s_p[t] = 1.0f / (1.0f + __expf(-s));
  }
  __syncthreads();
  if (t < L_ / 2) {
#pragma unroll
    for (int c = 0; c < C_; ++c) {
      const float p = s_p[c];
      bot[(size_t)b * (C_ * (L_ / 2)) + c * (L_ / 2) + t] =
          p * s_cm[c][(L_ / 2) + t] + (1.0f - p) * s_cm[c][t];
    }
  }
}

// ---------------------------------------------------------------------------
// Final classifier: logits = z @ Wc + bc  (N=200 is not a WMMA tile multiple;
// this GEMM is 0.2 GFLOP, so plain VALU is the right call).
// ---------------------------------------------------------------------------
__global__ __launch_bounds__(256) void logits_kernel(const float* __restrict__ z,
                                                     const float* __restrict__ Wc,
                                                     const float* __restrict__ bc,
                                                     float* __restrict__ out) {
  int i = blockIdx.x * 256 + threadIdx.x;
  if (i >= B_ * NCLS_) return;
  const int b = i / NCLS_;
  const int n = i - b * NCLS_;
  float s = bc[n];
  const float* zr = z + (size_t)b * HG_;
#pragma unroll 4
  for (int k = 0; k < HG_; ++k) s = fmaf(zr[k], Wc[k * NCLS_ + n], s);
  out[i] = s;
}

// ---------------------------------------------------------------------------
extern "C" void kernel_launch(void* const* d_in, const int* in_sizes, int n_in,
                              void* d_out, int out_size, void* d_ws, size_t ws_size,
                              hipStream_t stream) {
  const float* x       = (const float*)d_in[0];
  const float* fp      = (const float*)d_in[1];
  const float* mean    = (const float*)d_in[2];
  const float* Wg1     = (const float*)d_in[3];
  const float* bg1     = (const float*)d_in[4];
  const float* Wg2     = (const float*)d_in[5];
  const float* bg2     = (const float*)d_in[6];
  const float* Wf1     = (const float*)d_in[7];
  const float* bf1     = (const float*)d_in[8];
  const float* Wf2     = (const float*)d_in[9];
  const float* bf2     = (const float*)d_in[10];
  const float* Wg_head = (const float*)d_in[11];
  const float* bg_head = (const float*)d_in[12];
  const float* Wc      = (const float*)d_in[13];
  const float* bc      = (const float*)d_in[14];
  float* out = (float*)d_out;

  char* ws = (char*)d_ws;
  const size_t SZ_CM  = (size_t)B_ * C_ * L_ * sizeof(float);   // 32 MB
  const size_t SZ_SCR = (size_t)C_ * B_ * HF_ * sizeof(float);  // 128 MB
  float* cm      = (float*)ws;                  // (B, C, L)
  float* lat     = (float*)(ws + SZ_CM);        // (B, C, L)
  float* scratch = (float*)(ws + 2 * SZ_CM);    // reused: h -> hf -> bot+z
  float* h   = scratch;                                  // (C, B, H)   64 MB
  float* hf  = scratch;                                  // (C, B, HF) 128 MB
  float* bot = scratch;                                  // (B, 1024)   16 MB
  float* z   = scratch + (size_t)B_ * (C_ * (L_ / 2));   // (B, HG)      2 MB
  float* gate  = (float*)(ws + 2 * SZ_CM + SZ_SCR);      // (C, D)
  float* gmean = gate + C_ * D_;                         // (C, D)

  gate_kernel<<<(C_ * D_ + 255) / 256, 256, 0, stream>>>(fp, mean, gate, gmean);

  // h = relu(masked @ Wg1 + bg1)          -> (C, B, H)
  wmma_gemm<true><<<dim3(H_ / 128, B_ / 128, C_), 256, 0, stream>>>(
      x, 0, D_, Wg1, (long long)D_ * H_, H_, bg1, H_,
      h, (long long)B_ * H_, H_, D_, gate, gmean, 1);

  // cm = h @ Wg2 + bg2                    -> stored transposed as (B, C, L)
  wmma_gemm<false><<<dim3(L_ / 128, B_ / 128, C_), 256, 0, stream>>>(
      h, (long long)B_ * H_, H_, Wg2, (long long)H_ * L_, L_, bg2, L_,
      cm, L_, C_ * L_, H_, nullptr, nullptr, 0);

  // hf = relu(masked @ Wf1 + bf1)         -> (C, B, HF), reuses h region
  wmma_gemm<true><<<dim3(HF_ / 128, B_ / 128, C_), 256, 0, stream>>>(
      x, 0, D_, Wf1, 0, HF_, bf1, 0,
      hf, (long long)B_ * HF_, HF_, D_, gate, gmean, 1);

  // lat = hf @ Wf2 + bf2                  -> stored transposed as (B, C, L)
  wmma_gemm<false><<<dim3(L_ / 128, B_ / 128, C_), 256, 0, stream>>>(
      hf, (long long)B_ * HF_, HF_, Wf2, 0, L_, bf2, 0,
      lat, L_, C_ * L_, HF_, nullptr, nullptr, 0);

  // normalize / concept prob / bottleneck -> (B, 1024)
  fuse_kernel<<<B_, 128, 0, stream>>>(cm, lat, bot);

  // z = relu(bot @ Wg_head + bg_head)     -> (B, HG)
  wmma_gemm<false><<<dim3(HG_ / 128, B_ / 128, 1), 256, 0, stream>>>(
      bot, 0, C_ * (L_ / 2), Wg_head, 0, HG_, bg_head, 0,
      z, 0, HG_, C_ * (L_ / 2), nullptr, nullptr, 1);

  // logits = z @ Wc + bc                  -> d_out
  logits_kernel<<<(B_ * NCLS_ + 255) / 256, 256, 0, stream>>>(z, Wc, bc, out);
}